// Vox_fusion_net_32246614458676
// MI455X (gfx1250) — compile-verified
//
#include <hip/hip_runtime.h>
#include <math.h>
#include <stdint.h>

typedef float v2f __attribute__((ext_vector_type(2)));
typedef float v8f __attribute__((ext_vector_type(8)));

#define NCAMS   6
#define CFEAT   64
#define HH      88
#define WW      160
#define HWIMG   (HH*WW)
#define NVOX    200000
#define KPAD    68          // 65 channels padded to multiple of 4
#define VPRE    64
#define TILES   (NVOX/16)   // 12500, exact
#define WPB     4           // waves per block (wave32)
#define LDS_PER_WAVE (2*KPAD*16 + 32)   // f1[68][16], f2[68][16], cnt[32]
#define WLDS     (3*VPRE*KPAD)          // padded weights in LDS (13056 floats)
#define WBYTES   (WLDS*4)               // 52224 bytes
#define WCHUNKS  (WBYTES/16)            // 3264 b128 transfers

// ---------------------------------------------------------------------------
// Prep: split w_o into A/B halves and pad all K dims 65 -> 68 with zeros.
// Layout in d_ws: Wno_p[64][68] | Woa_p[64][68] | Wob_p[64][68]
// ---------------------------------------------------------------------------
__global__ void prep_weights(const float* __restrict__ w_no,
                             const float* __restrict__ w_o,
                             float* __restrict__ Wp) {
    int idx = blockIdx.x * 256 + threadIdx.x;
    const int total = 3 * VPRE * KPAD;
    if (idx >= total) return;
    int which = idx / (VPRE * KPAD);
    int rem   = idx % (VPRE * KPAD);
    int m = rem / KPAD, k = rem % KPAD;
    float v = 0.f;
    if (k < 65) {
        if (which == 0)      v = w_no[m * 65 + k];
        else if (which == 1) v = w_o[m * 130 + k];
        else                 v = w_o[m * 130 + 65 + k];
    }
    Wp[idx] = v;
}

// ---------------------------------------------------------------------------
// Fused voxel-fusion kernel. One wave32 = one 16-voxel tile:
//   phase 0: async-DMA the 52KB padded weight table into LDS (ASYNCcnt path)
//   phase 1: project 16 vox x 6 cams, bilinear-gather 65-ch features -> LDS
//   phase 2: two linear layers via V_WMMA_F32_16X16X4_F32 (A and B from LDS)
// ---------------------------------------------------------------------------
__global__ __launch_bounds__(WPB * 32) void vox_fusion(
    const float* __restrict__ feat,     // [6][64][88][160]
    const float* __restrict__ maskimg,  // [6][88][160]
    const float* __restrict__ Kmat,     // [6][4][4]
    const float* __restrict__ ext,      // [6][4][4]
    const float* __restrict__ Wp,       // [3][64][68] padded weights (global)
    const float* __restrict__ b_no,     // [64]
    const float* __restrict__ b_o,      // [64]
    float* __restrict__ out)            // [64][200000]
{
    __shared__ float smem[WLDS + WPB * LDS_PER_WAVE];

    const int lane = threadIdx.x & 31;
    const int wave = threadIdx.x >> 5;
    const int tile = blockIdx.x * WPB + wave;   // 3125*4 == 12500: no tail

    // ---- phase 0: async stage weights global -> LDS (whole workgroup) -------
    {
        // Low 32 bits of a generic LDS pointer == byte offset in LDS aperture.
        unsigned ldsW = (unsigned)(uintptr_t)(&smem[0]);
        for (int i = threadIdx.x; i < WCHUNKS; i += WPB * 32) {
            unsigned la = ldsW + (unsigned)i * 16u;
            unsigned go = (unsigned)i * 16u;
            asm volatile("global_load_async_to_lds_b128 %0, %1, %2"
                         :: "v"(la), "v"(go), "s"(Wp) : "memory");
        }
        asm volatile("s_wait_asynccnt 0" ::: "memory");
        __syncthreads();
    }

    const int v    = lane & 15;        // voxel within tile / N index
    const int buf  = lane >> 4;        // 0: cams {0,3,4} -> f1 ; 1: cams {1,2,5} -> f2
    const int nvox = tile * 16 + v;    // global voxel index

    float* wl   = smem + WLDS + wave * LDS_PER_WAVE;
    float* f1   = wl;                  // [68][16]
    float* f2   = wl + KPAD * 16;      // [68][16]
    float* cntl = wl + 2 * KPAD * 16;  // [32]

    // voxel position (layout z,y,x ; x fastest)
    int iz = nvox / 10000;
    int rr = nvox - iz * 10000;
    int iy = rr / 100;
    int ix = rr - iy * 100;
    float vx = -50.f + (float)ix;
    float vy = -50.f + (float)iy;
    float vz = -15.f + 1.5f * (float)iz;

    const int cams0[3] = {0, 3, 4};
    const int cams1[3] = {1, 2, 5};

    int   lin[3][4];
    float wgt[3][4];
    int   cbase[3];
    float depth_sum = 0.f;
    float cnt = 0.f;

    // ---- phase 1a: projections for this lane's 3 cameras --------------------
    #pragma unroll
    for (int j = 0; j < 3; ++j) {
        int cam = buf ? cams1[j] : cams0[j];
        const float* E  = ext  + cam * 16;
        const float* Km = Kmat + cam * 16;
        float lx = E[0]*vx + E[1]*vy + E[2]*vz  + E[3];
        float ly = E[4]*vx + E[5]*vy + E[6]*vz  + E[7];
        float lz = E[8]*vx + E[9]*vy + E[10]*vz + E[11];
        float cx = Km[0]*lx + Km[1]*ly + Km[2]*lz;
        float cy = Km[4]*lx + Km[5]*ly + Km[6]*lz;
        float cz = Km[8]*lx + Km[9]*ly + Km[10]*lz;
        float inv = 1.f / (cz + 1e-8f);
        float px = cx * inv, py = cy * inv;

        bool oob = (px < 0.f) || (px > (float)(WW-1)) ||
                   (py < 0.f) || (py > (float)(HH-1));

        // nearest-sample the occupancy mask
        float xr = roundf(px), yr = roundf(py);
        bool nv = (xr >= 0.f) && (xr <= (float)(WW-1)) &&
                  (yr >= 0.f) && (yr <= (float)(HH-1));
        int xi = (int)fminf(fmaxf(xr, 0.f), (float)(WW-1));
        int yi = (int)fminf(fmaxf(yr, 0.f), (float)(HH-1));
        float mval = maskimg[cam * HWIMG + yi * WW + xi];

        bool  valid = nv && (mval > 0.5f) && (lz > 0.f) && (!oob);
        float vm    = valid ? 1.f : 0.f;

        // bilinear taps (align_corners=True, zero padding)
        float x0f = floorf(px), y0f = floorf(py);
        float wx1 = px - x0f, wx0 = 1.f - wx1;
        float wy1 = py - y0f, wy0 = 1.f - wy1;
        float xs[2]  = {x0f, x0f + 1.f};
        float ys[2]  = {y0f, y0f + 1.f};
        float wxs[2] = {wx0, wx1};
        float wys[2] = {wy0, wy1};
        #pragma unroll
        for (int t = 0; t < 4; ++t) {
            float xf = xs[t & 1], yf = ys[t >> 1];
            bool tv = (xf >= 0.f) && (xf <= (float)(WW-1)) &&
                      (yf >= 0.f) && (yf <= (float)(HH-1));
            int xc = (int)fminf(fmaxf(xf, 0.f), (float)(WW-1));
            int yc = (int)fminf(fmaxf(yf, 0.f), (float)(HH-1));
            lin[j][t] = yc * WW + xc;
            wgt[j][t] = wxs[t & 1] * wys[t >> 1] * (tv ? 1.f : 0.f) * vm;
        }
        cbase[j] = cam * CFEAT * HWIMG;
        depth_sum += vm * lz * 0.01f;   // z / VOX_SIZE[0]
        cnt       += vm;
    }

    // ---- phase 1b: 65-ch gather-accumulate into this lane's LDS column ------
    float* fb = buf ? f2 : f1;
    for (int c = 0; c < CFEAT; ++c) {
        float s = 0.f;
        #pragma unroll
        for (int j = 0; j < 3; ++j) {
            const float* F = feat + cbase[j] + c * HWIMG;
            s += wgt[j][0] * F[lin[j][0]] + wgt[j][1] * F[lin[j][1]]
               + wgt[j][2] * F[lin[j][2]] + wgt[j][3] * F[lin[j][3]];
        }
        fb[c * 16 + v] = s;
    }
    fb[64 * 16 + v] = depth_sum;   // depth channel
    fb[65 * 16 + v] = 0.f;         // K padding
    fb[66 * 16 + v] = 0.f;
    fb[67 * 16 + v] = 0.f;
    cntl[lane] = cnt;

    // all tile producers/consumers are this same wave32: drain LDS stores
    asm volatile("s_wait_dscnt 0" ::: "memory");

    // ---- phase 2: WMMA f32 16x16x4 linear layers (A and B from LDS) ---------
    // A 16x4 layout: lanes 0-15 -> M=lane, K={0,1}; lanes 16-31 -> K={2,3}
    // B 4x16 layout: lanes 0-15 -> N=lane, K={0,1}; lanes 16-31 -> K={2,3}
    const int kh = buf ? 2 : 0;
    const int m  = lane & 15;
    const float* WnoL = smem;                     // [64][68] in LDS
    const float* WoaL = smem + VPRE * KPAD;
    const float* WobL = smem + 2 * VPRE * KPAD;

    v8f accN[4], accO[4];
    #pragma unroll
    for (int mt = 0; mt < 4; ++mt) {
        accN[mt] = (v8f){0.f,0.f,0.f,0.f,0.f,0.f,0.f,0.f};
        accO[mt] = (v8f){0.f,0.f,0.f,0.f,0.f,0.f,0.f,0.f};
    }

    for (int k0 = 0; k0 < KPAD; k0 += 4) {
        int kk = k0 + kh;
        v2f b1, b2, bn;
        b1.x = f1[kk * 16 + v];  b1.y = f1[(kk + 1) * 16 + v];
        b2.x = f2[kk * 16 + v];  b2.y = f2[(kk + 1) * 16 + v];
        bn.x = b1.x + b2.x;      bn.y = b1.y + b2.y;
        #pragma unroll
        for (int mt = 0; mt < 4; ++mt) {
            int row = mt * 16 + m;
            v2f aN = *(const v2f*)(WnoL + row * KPAD + kk);   // ds_load_b64
            v2f aA = *(const v2f*)(WoaL + row * KPAD + kk);
            v2f aB = *(const v2f*)(WobL + row * KPAD + kk);
            accN[mt] = __builtin_amdgcn_wmma_f32_16x16x4_f32(
                false, aN, false, bn, (short)0, accN[mt], false, false);
            accO[mt] = __builtin_amdgcn_wmma_f32_16x16x4_f32(
                false, aA, false, b1, (short)0, accO[mt], false, false);
            accO[mt] = __builtin_amdgcn_wmma_f32_16x16x4_f32(
                false, aB, false, b2, (short)0, accO[mt], false, false);
        }
    }

    // ---- epilogue: bias + ELU + count masks + store -------------------------
    float cntv = cntl[v] + cntl[16 + v];
    float mNo  = (cntv == 1.f) ? 1.f : 0.f;
    float mO   = (cntv == 2.f) ? 1.f : 0.f;
    const int mhalf = buf ? 8 : 0;   // C/D: lanes 16-31 hold M = 8+r

    #pragma unroll
    for (int mt = 0; mt < 4; ++mt) {
        #pragma unroll
        for (int r = 0; r < 8; ++r) {
            int mrow = mt * 16 + mhalf + r;
            float xn = accN[mt][r] + b_no[mrow];
            float en = xn > 0.f ? xn : (__expf(xn) - 1.f);
            float xo = accO[mt][r] + b_o[mrow];
            float eo = xo > 0.f ? xo : (__expf(xo) - 1.f);
            out[mrow * NVOX + (tile * 16 + v)] = en * mNo + eo * mO;
        }
    }
}

// ---------------------------------------------------------------------------
extern "C" void kernel_launch(void* const* d_in, const int* in_sizes, int n_in,
                              void* d_out, int out_size, void* d_ws, size_t ws_size,
                              hipStream_t stream) {
    (void)in_sizes; (void)n_in; (void)out_size; (void)ws_size;
    const float* feat    = (const float*)d_in[0];  // [1,6,64,88,160]
    const float* maskimg = (const float*)d_in[1];  // [1,6,1,88,160]
    const float* Kmat    = (const float*)d_in[2];  // [1,6,4,4]
    const float* ext     = (const float*)d_in[3];  // [1,6,4,4]
    const float* w_no    = (const float*)d_in[4];  // [64,65]
    const float* b_no    = (const float*)d_in[5];  // [64]
    const float* w_o     = (const float*)d_in[6];  // [64,130]
    const float* b_o     = (const float*)d_in[7];  // [64]
    float* out = (float*)d_out;                    // [1,64,200000]
    float* Wp  = (float*)d_ws;                     // 3*64*68 floats = 52 KB

    const int totalW = 3 * VPRE * KPAD;
    prep_weights<<<(totalW + 255) / 256, 256, 0, stream>>>(w_no, w_o, Wp);

    const int blocks = (TILES + WPB - 1) / WPB;    // 3125, exact
    vox_fusion<<<blocks, WPB * 32, 0, stream>>>(feat, maskimg, Kmat, ext,
                                                Wp, b_no, b_o, out);
}